// MultiHeadSelfAttention_68848325754951
// MI455X (gfx1250) — compile-verified
//
#include <hip/hip_runtime.h>
#include <stdint.h>

// Problem constants (from reference): B=4, S=2048, D=512, H=8, DEPTH=64
#define CB 4
#define CS 2048
#define CD 512
#define CH 8
#define CDEPTH 64
#define CM (CB * CS)      // 8192 rows of x

typedef float v2f __attribute__((ext_vector_type(2)));
typedef float v4f __attribute__((ext_vector_type(4)));
typedef float v8f __attribute__((ext_vector_type(8)));
typedef unsigned int u32x4 __attribute__((ext_vector_type(4)));
typedef int i32x4 __attribute__((ext_vector_type(4)));
typedef int i32x8 __attribute__((ext_vector_type(8)));

// F32 WMMA: D(16x16) = A(16x4) * B(4x16) + C.  Exact f32 matrix path on CDNA5.
__device__ __forceinline__ v8f wmma_f32(v2f a, v2f b, v8f c) {
    return __builtin_amdgcn_wmma_f32_16x16x4_f32(
        /*neg_a=*/false, a, /*neg_b=*/false, b,
        /*c_mod=*/(short)0, c, /*reuse_a=*/false, /*reuse_b=*/false);
}

// Per-lane async copy 16B global -> LDS (ASYNCcnt path, no VGPR round trip).
__device__ __forceinline__ void async_b128(uint32_t lds_off, const float* gptr) {
    asm volatile("global_load_async_to_lds_b128 %0, %1, off"
                 :: "v"(lds_off), "v"((uint64_t)(uintptr_t)gptr)
                 : "memory");
}
__device__ __forceinline__ void async_wait() {
    asm volatile("s_wait_asynccnt 0x0" ::: "memory");
}

// TDM: DMA a 128x64 f32 tile (row stride 512 elems) into LDS, padding LDS by
// 4 dwords after every 64 dwords -> LDS row stride 68 floats (conflict-free).
// D# built per CDNA5 ISA ch.8: group0 = {flags, lds_addr, global_addr, type},
// group1 = {data_size/pad, tensor dims, tile dims, strides}.
__device__ __forceinline__ void tdm_load_tile_128x64(uint32_t lds_off, const float* gsrc) {
    const uint64_t ga = (uint64_t)(uintptr_t)gsrc;
    u32x4 g0;
    g0[0] = 1u;                                   // count=1 (valid descriptor)
    g0[1] = lds_off;                              // lds_addr (bytes)
    g0[2] = (uint32_t)ga;                         // global_addr[31:0]
    g0[3] = (uint32_t)(ga >> 32) | (2u << 30);    // global_addr[56:32] | type=2
    i32x8 g1;
    g1[0] = (2 << 16)      // data_size = 4B
          | (1 << 20)      // pad_enable
          | (5 << 22)      // pad_interval: 64 dwords
          | (3 << 25);     // pad_amount: 4 dwords
    g1[1] = (64  << 16);   // tensor_dim0 = 64 (bits 79:48)
    g1[2] = (128 << 16);   // tensor_dim1 = 128 (bits 111:80)
    g1[3] = (64  << 16);   // tile_dim0 = 64 (bits 127:112)
    g1[4] = 128;           // tile_dim1 = 128 (bits 143:128); tile_dim2 = 0
    g1[5] = CD;            // tensor_dim0_stride = 512 elems (bits 207:160)
    g1[6] = 0;
    g1[7] = 0;
    const i32x4 z4 = {0, 0, 0, 0};                // groups 2/3 unused (2D tile)
    const i32x8 z8 = {0, 0, 0, 0, 0, 0, 0, 0};
    __builtin_amdgcn_tensor_load_to_lds(g0, g1, z4, z4, z8, 0);
}

// ---------------------------------------------------------------------------
// Tiled GEMM with bias:  C[M,512] = A[M,512] @ W[512,512] + bias[512]
// Block: 256 threads = 8 waves (2x4 grid of 16x16 WMMA tiles) -> 32x64 macro
// tile.  Staging uses global_load_async_to_lds_b128 (ASYNCcnt).
// ---------------------------------------------------------------------------
#define LPAD 4
__global__ __launch_bounds__(256)
void gemm_bias_kernel(const float* __restrict__ A, const float* __restrict__ W,
                      const float* __restrict__ bias, float* __restrict__ C) {
    __shared__ float lA[32][64 + LPAD];
    __shared__ float lB[64][64 + LPAD];

    const int t    = threadIdx.x;
    const int wave = t >> 5;
    const int lane = t & 31;
    const int wr   = wave >> 2;       // 0..1
    const int wc   = wave & 3;        // 0..3
    const int r16  = lane & 15;       // 0..15
    const int hi   = lane >> 4;       // 0/1 -> K pair select / M-half select
    const int m0   = blockIdx.x * 32;
    const int n0   = blockIdx.y * 64;

    v8f acc = {};

    for (int kt = 0; kt < CD; kt += 64) {
        // Stage A tile (32x64): 512 x b128, 2 per lane, async into LDS
        #pragma unroll
        for (int i = 0; i < 2; ++i) {
            int idx = t + i * 256;
            int r = idx >> 4;              // 16 float4 per row
            int c4 = (idx & 15) << 2;
            async_b128((uint32_t)(uintptr_t)&lA[r][c4],
                       A + (size_t)(m0 + r) * CD + kt + c4);
        }
        // Stage B tile (64x64): 1024 x b128, 4 per lane, async into LDS
        #pragma unroll
        for (int i = 0; i < 4; ++i) {
            int idx = t + i * 256;
            int r = idx >> 4;
            int c4 = (idx & 15) << 2;
            async_b128((uint32_t)(uintptr_t)&lB[r][c4],
                       W + (size_t)(kt + r) * CD + n0 + c4);
        }
        async_wait();
        __syncthreads();

        #pragma unroll
        for (int kk = 0; kk < 64; kk += 4) {
            const int kb = kk + (hi << 1);  // lanes 0-15: K=kk,kk+1; 16-31: +2,+3
            v2f a, b;
            a.x = lA[wr * 16 + r16][kb];
            a.y = lA[wr * 16 + r16][kb + 1];
            b.x = lB[kb][wc * 16 + r16];
            b.y = lB[kb + 1][wc * 16 + r16];
            acc = wmma_f32(a, b, acc);
        }
        __syncthreads();
    }

    // C/D layout: VGPR i -> row (i + hi*8), col r16
    #pragma unroll
    for (int i = 0; i < 8; ++i) {
        int r = m0 + wr * 16 + i + hi * 8;
        int c = n0 + wc * 16 + r16;
        C[(size_t)r * CD + c] = acc[i] + bias[c];
    }
}

// ---------------------------------------------------------------------------
// Fused attention for one (b, h, 16-query tile):
//   logits(16x2048) = Q @ K^T * 0.125   (WMMA; K tiles DMA'd by TDM)
//   softmax rows in LDS, write attn once (coalesced)
//   ctx(16x64) = P @ V                  (WMMA; V tiles DMA'd by TDM; k-split
//                                        across waves, LDS reduce)
// Dynamic LDS: lQ 16x68 | lKV 128x68 (reused as reduce scratch) | lP 16x2052
// ---------------------------------------------------------------------------
#define PSTR 2052
__global__ __launch_bounds__(256)
void attn_fused_kernel(const float* __restrict__ Q, const float* __restrict__ K,
                       const float* __restrict__ V, float* __restrict__ attn,
                       float* __restrict__ ctx) {
    extern __shared__ float smem[];
    float* lQ  = smem;                      // 16 * 68   = 1088
    float* lKV = lQ + 16 * (64 + LPAD);     // 128 * 68  = 8704
    float* lP  = lKV + 128 * (64 + LPAD);   // 16 * 2052 = 32832
    float* lR  = lKV;                       // reduce scratch (8*16*16) reuses lKV

    const int t    = threadIdx.x;
    const int wave = t >> 5;
    const int lane = t & 31;
    const int r16  = lane & 15;
    const int hi   = lane >> 4;

    const int b  = blockIdx.z;
    const int h  = blockIdx.y;
    const int q0 = blockIdx.x * 16;

    const size_t headOff = (size_t)b * CS * CD + (size_t)h * CDEPTH;
    const float* Qp = Q + headOff + (size_t)q0 * CD;
    const float* Kp = K + headOff;
    const float* Vp = V + headOff;

    const uint32_t lKVoff = (uint32_t)(uintptr_t)lKV;

    // Async-copy Q tile (16x64) raw; scale folded into logits store.
    {
        int r = t >> 4;
        int c4 = (t & 15) << 2;
        async_b128((uint32_t)(uintptr_t)&lQ[r * (64 + LPAD) + c4],
                   Qp + (size_t)r * CD + c4);
    }
    async_wait();

    // ---- logits: per 128-key stage, TDM the K tile, each wave one 16x16 ----
    for (int kt = 0; kt < CS; kt += 128) {
        __syncthreads();   // publish lQ (first iter) / retire lKV reads (later)
        if (wave == 0) {
            tdm_load_tile_128x64(lKVoff, Kp + (size_t)kt * CD);
            __builtin_amdgcn_s_wait_tensorcnt(0);
        }
        __syncthreads();

        v8f c = {};
        #pragma unroll
        for (int d = 0; d < CDEPTH; d += 4) {
            const int kb = d + (hi << 1);
            v2f a, bf;
            a.x  = lQ[r16 * (64 + LPAD) + kb];
            a.y  = lQ[r16 * (64 + LPAD) + kb + 1];
            // B(4x16) = K^T : element [kb][key] = Ktile[key][kb]
            bf.x = lKV[(wave * 16 + r16) * (64 + LPAD) + kb];
            bf.y = lKV[(wave * 16 + r16) * (64 + LPAD) + kb + 1];
            c = wmma_f32(a, bf, c);
        }
        #pragma unroll
        for (int i = 0; i < 8; ++i)   // fold 1/sqrt(depth)=0.125 here
            lP[(i + hi * 8) * PSTR + kt + wave * 16 + r16] = c[i] * 0.125f;
    }
    __syncthreads();

    // ---- softmax over 2048 keys; wave handles rows 2*wave, 2*wave+1 ----
    #pragma unroll
    for (int rr = 0; rr < 2; ++rr) {
        const int r = wave * 2 + rr;
        float* prow = lP + r * PSTR;
        float m = -3.0e38f;
        for (int j = lane; j < CS; j += 32) m = fmaxf(m, prow[j]);
        #pragma unroll
        for (int s = 16; s > 0; s >>= 1) m = fmaxf(m, __shfl_xor(m, s, 32));
        float sum = 0.f;
        for (int j = lane; j < CS; j += 32) {
            float e = __expf(prow[j] - m);
            prow[j] = e;
            sum += e;
        }
        #pragma unroll
        for (int s = 16; s > 0; s >>= 1) sum += __shfl_xor(sum, s, 32);
        const float inv = 1.0f / sum;
        float* arow = attn + (((size_t)(b * CH + h) * CS) + (q0 + r)) * CS;
        for (int j = lane; j < CS; j += 32) {   // coalesced attn store
            float p = prow[j] * inv;
            prow[j] = p;
            arow[j] = p;
        }
    }

    // ---- ctx = P(16x2048) @ V(2048x64): TDM V tiles; wave=(khalf,ntile) ----
    {
        const int ntile = wave & 3;
        const int kh    = wave >> 2;
        v8f c = {};
        for (int kc = 0; kc < CS; kc += 128) {
            __syncthreads();   // retire prior lKV reads (and lP writes, 1st iter)
            if (wave == 0) {
                tdm_load_tile_128x64(lKVoff, Vp + (size_t)kc * CD);
                __builtin_amdgcn_s_wait_tensorcnt(0);
            }
            __syncthreads();
            #pragma unroll
            for (int k = 0; k < 64; k += 4) {
                const int kl = kh * 64 + k + (hi << 1);   // row in V tile
                const int kg = kc + kl;                   // column in P
                v2f a, bf;
                a.x  = lP[r16 * PSTR + kg];
                a.y  = lP[r16 * PSTR + kg + 1];
                bf.x = lKV[kl * (64 + LPAD) + ntile * 16 + r16];
                bf.y = lKV[(kl + 1) * (64 + LPAD) + ntile * 16 + r16];
                c = wmma_f32(a, bf, c);
            }
        }
        __syncthreads();
        // cross-wave reduce over khalf through LDS (reuses lKV region)
        #pragma unroll
        for (int i = 0; i < 8; ++i)
            lR[(wave * 16 + i + hi * 8) * 16 + r16] = c[i];
        __syncthreads();
        if (kh == 0) {
            #pragma unroll
            for (int i = 0; i < 8; ++i) {
                const int rloc = i + hi * 8;
                float s = lR[(ntile * 16 + rloc) * 16 + r16] +
                          lR[((ntile + 4) * 16 + rloc) * 16 + r16];
                ctx[(size_t)(b * CS + q0 + rloc) * CD + h * CDEPTH + ntile * 16 + r16] = s;
            }
        }
    }
}

// ---------------------------------------------------------------------------
extern "C" void kernel_launch(void* const* d_in, const int* in_sizes, int n_in,
                              void* d_out, int out_size, void* d_ws, size_t ws_size,
                              hipStream_t stream) {
    (void)in_sizes; (void)n_in; (void)out_size; (void)ws_size;

    const float* x  = (const float*)d_in[0];
    const float* Wq = (const float*)d_in[1];
    const float* bq = (const float*)d_in[2];
    const float* Wk = (const float*)d_in[3];
    const float* bk = (const float*)d_in[4];
    const float* Wv = (const float*)d_in[5];
    const float* bv = (const float*)d_in[6];
    const float* Wo = (const float*)d_in[7];
    const float* bo = (const float*)d_in[8];

    float* out  = (float*)d_out;                       // [B,S,D]
    float* attn = out + (size_t)CB * CS * CD;          // [B,H,S,S]

    float* Q  = (float*)d_ws;                          // [M,D]
    float* K  = Q  + (size_t)CM * CD;
    float* V  = K  + (size_t)CM * CD;
    float* CT = V  + (size_t)CM * CD;                  // ctx [M,D]

    dim3 gG(CM / 32, CD / 64);   // 256 x 8
    gemm_bias_kernel<<<gG, 256, 0, stream>>>(x, Wq, bq, Q);
    gemm_bias_kernel<<<gG, 256, 0, stream>>>(x, Wk, bk, K);
    gemm_bias_kernel<<<gG, 256, 0, stream>>>(x, Wv, bv, V);

    const size_t smemBytes =
        (size_t)(16 * (64 + LPAD) + 128 * (64 + LPAD) + 16 * PSTR) * sizeof(float);
    (void)hipFuncSetAttribute((const void*)attn_fused_kernel,
                              hipFuncAttributeMaxDynamicSharedMemorySize,
                              (int)smemBytes);
    dim3 gA(CS / 16, CH, CB);    // 128 x 8 x 4
    attn_fused_kernel<<<gA, 256, smemBytes, stream>>>(Q, K, V, attn, CT);

    gemm_bias_kernel<<<gG, 256, 0, stream>>>(CT, Wo, bo, out);
}